// ProposalTargetLayer_88115549045405
// MI455X (gfx1250) — compile-verified
//
#include <hip/hip_runtime.h>
#include <hip/hip_bf16.h>
#include <math.h>

// ---------------------------------------------------------------------------
// ProposalTargetLayer for gfx1250 (MI455X).
// d^2(r,g) = |r|^2 + |g|^2 - 2*(rx*gx + ry*gy); cross term via
// V_WMMA_F32_16X16X4_F32 (K padded 2->4 with zeros, full f32 precision).
// All threshold tests done on squared distances (thresholds' squares are
// exactly representable); per-GT argmax is a branchless packed-key u64 max.
// ---------------------------------------------------------------------------

#define B_ 8
#define R_ 4096
#define G_ 512
#define OUTG_ (G_ * 35)
#define OUTR_ (R_ * 2)
#define OUTB_ (OUTG_ + OUTR_)

typedef float v2f __attribute__((ext_vector_type(2)));
typedef float v8f __attribute__((ext_vector_type(8)));

__constant__ int   c_GROUP[10] = {0, 1, 1, 2, 2, 3, 4, 4, 5, 5};
// squared thresholds (exact): HIGH_T^2, MED_T^2
__constant__ float c_HIGH2[10] = {4.f, 6.25f, 6.25f, 12.25f, 6.25f, 0.25f, 1.f, 1.f, 0.25f, 0.25f};
__constant__ float c_MED2[10]  = {16.f, 16.f, 16.f, 16.f, 16.f, 6.25f, 6.25f, 6.25f, 4.f, 4.f};

__device__ __forceinline__ int clamp09(int v) {
    return v < 0 ? 0 : (v > 9 ? 9 : v);
}

__device__ __forceinline__ unsigned long long shflxor16_u64(unsigned long long v) {
    const int lo = __shfl_xor((int)(unsigned)v, 16, 32);
    const int hi = __shfl_xor((int)(unsigned)(v >> 32), 16, 32);
    return ((unsigned long long)(unsigned)hi << 32) | (unsigned)lo;
}

// One wave (32 threads) handles a tile of 16 GTs for one batch; loops over all
// ROIs in chunks of 16, computing the 16x16 cross-product tile with WMMA.
__global__ __launch_bounds__(32)
void match_kernel(const float* __restrict__ rois,     // (B,R,9)
                  const float* __restrict__ scores,   // (B,R)
                  const float* __restrict__ gt,       // (B,G,10)
                  const int*   __restrict__ labels,   // (B,R)
                  float* __restrict__ out) {
    const int lane = threadIdx.x;          // 0..31, wave32
    const int tile = blockIdx.x & ((G_ / 16) - 1);
    const int b    = blockIdx.x / (G_ / 16);
    const int g    = tile * 16 + (lane & 15);

    // ---- per-lane GT data (upper half lanes duplicate for rows 8..15) ----
    const float* gtrow = gt + (size_t)(b * G_ + g) * 10;
    float gv[10];
    float gsum = 0.f;
#pragma unroll
    for (int k = 0; k < 10; ++k) { gv[k] = gtrow[k]; gsum += gv[k]; }
    const bool  validGt = (gsum != 0.f);
    const float gx = gv[0], gy = gv[1];
    const int   cls  = clamp09((int)gv[9]);
    const float ht2  = c_HIGH2[cls];
    const float mt2  = c_MED2[cls];
    const float gn   = gx * gx + gy * gy;
    // single-compare match key: roi must have  grp|valid  ==  ggrp|0x100
    const unsigned ggrpKey =
        validGt ? (unsigned)(c_GROUP[cls] | 0x100) : 0xFFFFFFFFu;

    // B matrix 4x16: lanes 0-15 hold K=0,1 (x,y); lanes 16-31 hold K=2,3 = 0
    v2f Bm;
    Bm.x = (lane < 16) ? gx : 0.f;
    Bm.y = (lane < 16) ? gy : 0.f;

    // staged per-ROI row: (|r|^2 bits, group|valid<<8, score bits, ~idx)
    __shared__ uint4 lroi[16];

    unsigned long long bestH = 0ull, bestM = 0ull;

    const float* roiB = rois   + (size_t)b * R_ * 9;
    const float* scB  = scores + (size_t)b * R_;
    const int*   lbB  = labels + (size_t)b * R_;

    for (int r0 = 0; r0 < R_; r0 += 16) {
        const int i = r0 + (lane & 15);
        const float* rr = roiB + (size_t)i * 9;
        float rvv[9];
        float rsum = 0.f;
#pragma unroll
        for (int k = 0; k < 9; ++k) { rvv[k] = rr[k]; rsum += rvv[k]; }
        const float rx = rvv[0], ry = rvv[1];
        const int   lb = clamp09(lbB[i]);
        const unsigned mk =
            (unsigned)c_GROUP[lb] | ((rsum != 0.f) ? 0x100u : 0u);
        const float sc = scB[i];
        const float rn = rx * rx + ry * ry;

        // both wave halves write identical values -> benign duplicate store
        lroi[lane & 15] =
            make_uint4(__float_as_uint(rn), mk, __float_as_uint(sc),
                       ~(unsigned)i);

        // A matrix 16x4: lanes 0-15 hold K=0,1 (x,y); lanes 16-31 K=2,3 = 0
        v2f Am;
        Am.x = (lane < 16) ? rx : 0.f;
        Am.y = (lane < 16) ? ry : 0.f;

        __syncthreads();  // single-wave WG: barrier->S_NOP, but orders LDS

        v8f c = {};
        // cross[m][n] = rx[m]*gx[n] + ry[m]*gy[n]   (f32 WMMA, EXEC all-1s)
        c = __builtin_amdgcn_wmma_f32_16x16x4_f32(
                /*neg_a=*/false, Am, /*neg_b=*/false, Bm,
                /*c_mod=*/(short)0, c, /*reuse_a=*/false, /*reuse_b=*/false);

#pragma unroll
        for (int v = 0; v < 8; ++v) {
            const int m = v + ((lane >> 4) << 3);  // rows 0..7 / 8..15
            const uint4 w = lroi[m];
            const float d2 = fmaf(-2.f, c[v], __uint_as_float(w.x) + gn);
            const bool ok   = (w.y == ggrpKey);
            const bool inHi = ok && (d2 <= ht2);
            const bool inMe = ok && (d2 > ht2) && (d2 <= mt2);
            const unsigned long long key =
                ((unsigned long long)w.z << 32) | w.w;
            bestH = (inHi && key > bestH) ? key : bestH;   // branchless
            bestM = (inMe && key > bestM) ? key : bestM;
        }
        __syncthreads();
    }

    // ---- combine the two half-wave partials for each GT (rows 0-7 / 8-15) --
    {
        const unsigned long long oH = shflxor16_u64(bestH);
        const unsigned long long oM = shflxor16_u64(bestM);
        bestH = (oH > bestH) ? oH : bestH;
        bestM = (oM > bestM) ? oM : bestM;
    }

    if (lane < 16) {
        const bool hasHigh   = bestH != 0ull;
        const bool hasMed    = bestM != 0ull;
        const bool selMed    = !hasHigh && hasMed;
        const bool matched   = hasHigh || selMed;
        const bool unmatched = validGt && !hasHigh && !hasMed;

        const float bestHS = __uint_as_float((unsigned)(bestH >> 32));
        const int   bestHI = (int)~(unsigned)bestH;
        const float bestMS = __uint_as_float((unsigned)(bestM >> 32));
        const int   bestMI = (int)~(unsigned)bestM;

        float* o = out + (size_t)b * OUTB_ + (size_t)g * 35;

        if (hasHigh) {
            const float* hr = roiB + (size_t)bestHI * 9;
#pragma unroll
            for (int k = 0; k < 9; ++k) o[k] = hr[k];
            o[9]  = fminf((1.2f + bestHS) * 0.5f, 1.0f);
            o[10] = 1.f;
        } else {
#pragma unroll
            for (int k = 0; k < 11; ++k) o[k] = 0.f;
        }

        if (selMed) {
            const float* mr = roiB + (size_t)bestMI * 9;
#pragma unroll
            for (int k = 0; k < 9; ++k) o[11 + k] = mr[k];
            o[20] = bestMS;
            o[21] = 1.f;
        } else {
#pragma unroll
            for (int k = 0; k < 11; ++k) o[11 + k] = 0.f;
        }

#pragma unroll
        for (int k = 0; k < 10; ++k) o[22 + k] = matched ? gv[k] : 0.f;

        const float rng  = sqrtf(gn);
        const bool nearb = unmatched && (rng < 30.f);
        const bool farb  = unmatched && (rng > 50.f);
        const bool midb  = unmatched && !nearb && !farb;
        o[32] = nearb ? 1.f : 0.f;
        o[33] = midb  ? 1.f : 0.f;
        o[34] = farb  ? 1.f : 0.f;
    }
}

// Precompute (gx, gy) per GT; invalid GTs get x=1e30 so their distance is
// effectively +inf and the per-ROI scan needs no per-iteration select.
__global__ __launch_bounds__(256)
void gtinfo_kernel(const float* __restrict__ gt, float2* __restrict__ info) {
    const int t = blockIdx.x * blockDim.x + threadIdx.x;
    if (t >= B_ * G_) return;
    const float* g = gt + (size_t)t * 10;
    float s = 0.f;
#pragma unroll
    for (int k = 0; k < 10; ++k) s += g[k];
    const bool valid = (s != 0.f);
    info[t] = make_float2(valid ? g[0] : 1e30f, valid ? g[1] : 0.f);
}

// One thread per ROI: min squared distance over all GTs (staged in LDS).
// fp <=> dmin > 4.0 <=> dmin^2 > 16 (exact threshold square).
__global__ __launch_bounds__(256)
void fp_kernel(const float* __restrict__ rois,
               const float* __restrict__ scores,
               const float2* __restrict__ info,
               float* __restrict__ out) {
    __shared__ float2 sg[G_];
    const int b = blockIdx.x / (R_ / 256);
    const int r = (blockIdx.x % (R_ / 256)) * 256 + threadIdx.x;

    for (int k = threadIdx.x; k < G_; k += 256) sg[k] = info[b * G_ + k];
    __syncthreads();

    const float* rr = rois + ((size_t)b * R_ + r) * 9;
    float s = 0.f, rv0 = 0.f, rv1 = 0.f;
#pragma unroll
    for (int k = 0; k < 9; ++k) {
        const float v = rr[k];
        if (k == 0) rv0 = v;
        if (k == 1) rv1 = v;
        s += v;
    }
    const bool validRoi = (s != 0.f);

    float dmin2 = INFINITY;
    for (int g = 0; g < G_; ++g) {
        const float2 gi = sg[g];
        const float dx = rv0 - gi.x;
        const float dy = rv1 - gi.y;
        dmin2 = fminf(dmin2, fmaf(dx, dx, dy * dy));
    }
    const bool fp = validRoi && (dmin2 > 16.0f);
    const float sc = scores[(size_t)b * R_ + r];

    float* o = out + (size_t)b * OUTB_ + OUTG_ + (size_t)r * 2;
    o[0] = fp ? sc : 0.f;
    o[1] = fp ? 1.f : 0.f;
}

extern "C" void kernel_launch(void* const* d_in, const int* in_sizes, int n_in,
                              void* d_out, int out_size, void* d_ws, size_t ws_size,
                              hipStream_t stream) {
    const float* rois       = (const float*)d_in[0];  // (B,R,9)
    const float* roi_scores = (const float*)d_in[1];  // (B,R)
    const float* gt         = (const float*)d_in[2];  // (B,G,10)
    const int*   roi_labels = (const int*)d_in[3];    // (B,R)
    // d_in[4] = batch_size scalar (constants baked in)

    float*  out  = (float*)d_out;
    float2* info = (float2*)d_ws;                     // B*G*8 = 32 KB scratch

    gtinfo_kernel<<<(B_ * G_ + 255) / 256, 256, 0, stream>>>(gt, info);
    match_kernel<<<B_ * (G_ / 16), 32, 0, stream>>>(rois, roi_scores, gt,
                                                    roi_labels, out);
    fp_kernel<<<B_ * (R_ / 256), 256, 0, stream>>>(rois, roi_scores, info, out);
}